// SpatialGrid2D_21234318312197
// MI455X (gfx1250) — compile-verified
//
#include <hip/hip_runtime.h>

// Bilinear table interpolation: out[p][c] = bilerp(table, uv[p])[c]
// Table: 1024 x 1024 x 8 f32 (32 MB, L2-resident on MI455X's 192 MB L2).
// HBM-bound: ~160 MB of stream traffic -> ~7us floor at 23.3 TB/s.

typedef float v2f __attribute__((ext_vector_type(2)));
typedef float v4f __attribute__((ext_vector_type(4)));

#define TABLE_H 1024
#define TABLE_W 1024
#define TABLE_C 8

__global__ __launch_bounds__(256) void SpatialGrid2D_bilerp_kernel(
    const float* __restrict__ uvList,
    const float* __restrict__ table,
    float* __restrict__ out,
    int n)
{
    int i = blockIdx.x * blockDim.x + threadIdx.x;
    if (i >= n) return;

    // uv stream: read exactly once -> non-temporal load (TH_LOAD_NT),
    // keeps these lines from evicting the L2-resident table.
    v2f uv = __builtin_nontemporal_load((const v2f*)uvList + i);

    // Match reference: indexFloat = uv * [W-1, H-1]; trunc to int; frac.
    float fx = uv.x * (float)(TABLE_W - 1);
    float fy = uv.y * (float)(TABLE_H - 1);
    int ix = (int)fx;            // uv in [0,1): ix <= W-2, iy <= H-2
    int iy = (int)fy;
    float alpha = fx - (float)ix;
    float beta  = fy - (float)iy;

    // Texels (iy,ix) and (iy,ix+1) are 64 contiguous bytes (C==8 floats each).
    // Row pair iy   : a[0..7], b[0..7]   -> 4x global_load_b128
    // Row pair iy+1 : c[0..7], d[0..7]   -> 4x global_load_b128
    // Default (RT) temporal hint: table gathers should hit + stay in L2.
    const v4f* r0 = (const v4f*)(table + ((size_t)iy * TABLE_W + ix) * TABLE_C);
    const v4f* r1 = r0 + (TABLE_W * TABLE_C / 4);   // +1 row in y

    v4f a0 = r0[0], a1 = r0[1];   // table[iy  ][ix  ][0..7]
    v4f b0 = r0[2], b1 = r0[3];   // table[iy  ][ix+1][0..7]
    v4f c0 = r1[0], c1 = r1[1];   // table[iy+1][ix  ][0..7]
    v4f d0 = r1[2], d1 = r1[3];   // table[iy+1][ix+1][0..7]

    // Same algebraic form as the reference: x*(1-t) + t*y
    float ia = 1.0f - alpha;
    float ib = 1.0f - beta;

    v4f p0 = a0 * ia + alpha * b0;
    v4f p1 = a1 * ia + alpha * b1;
    v4f q0 = c0 * ia + alpha * d0;
    v4f q1 = c1 * ia + alpha * d1;

    v4f o0 = p0 * ib + beta * q0;
    v4f o1 = p1 * ib + beta * q1;

    // Output stream: write-once 128 MB -> non-temporal stores (TH_STORE_NT)
    // so the store stream does not flush the 32 MB table out of L2.
    v4f* dst = (v4f*)out + (size_t)i * 2;
    __builtin_nontemporal_store(o0, dst);
    __builtin_nontemporal_store(o1, dst + 1);
}

extern "C" void kernel_launch(void* const* d_in, const int* in_sizes, int n_in,
                              void* d_out, int out_size, void* d_ws, size_t ws_size,
                              hipStream_t stream) {
    const float* uvList = (const float*)d_in[0];   // (N, 2) f32
    const float* table  = (const float*)d_in[1];   // (1024, 1024, 8) f32
    float* out = (float*)d_out;                    // (N, 8) f32

    int n = in_sizes[0] / 2;                       // N points
    const int threads = 256;                       // 8 wave32s per block
    int blocks = (n + threads - 1) / threads;

    SpatialGrid2D_bilerp_kernel<<<blocks, threads, 0, stream>>>(uvList, table, out, n);
}